// MHPFGT_33517924778184
// MI455X (gfx1250) — compile-verified
//
#include <hip/hip_runtime.h>

typedef __attribute__((ext_vector_type(2))) float v2f;
typedef __attribute__((ext_vector_type(8))) float v8f;

#define NNODES 20000
#define NEDGES 160000
#define STATE  704   // 640 (M) + 64 (Kt) floats per node

static_assert(NNODES % 16 == 0, "row tiling assumes multiple of 16");
static_assert(STATE % 4 == 0, "float4 scatter");

__device__ __forceinline__ void atomicAddF(float* p, float v) {
    __hip_atomic_fetch_add(p, v, __ATOMIC_RELAXED, __HIP_MEMORY_SCOPE_AGENT);
}

// ---------------- degree / normalization ----------------
__global__ void k_fill(float* p, float v, int n) {
    int i = blockIdx.x * blockDim.x + threadIdx.x;
    if (i < n) p[i] = v;
}
__global__ void k_count(const int* __restrict__ col, float* __restrict__ deg, int e) {
    int i = blockIdx.x * blockDim.x + threadIdx.x;
    if (i < e) atomicAddF(&deg[col[i]], 1.0f);
}
__global__ void k_rsqrt(float* p, int n) {
    int i = blockIdx.x * blockDim.x + threadIdx.x;
    if (i < n) p[i] = rsqrtf(p[i]);   // deg >= 1 always (self loop)
}

// ---------------- generic f32 WMMA GEMM (compile-time shapes) -----------------
// out[r,c] = epi( sum_k A[r,k]*W[k,c] + bias[c] ),  A: nrows x KDIM, W: KDIM x NCOLS
// Block = 16 rows; wave w covers columns [16w, 16w+16). EPI: 0=id, 1=relu, 2=1+elu
template<int KDIM, int NCOLS, int EPI>
__global__ void k_gemm_wmma(const float* __restrict__ A, const float* __restrict__ W,
                            const float* __restrict__ bias, float* __restrict__ out)
{
    constexpr int NCP  = (NCOLS + 15) & ~15;        // padded cols (zero-filled)
    constexpr int KC   = (KDIM < 64) ? KDIM : 64;   // K chunk held in LDS
    constexpr int ASTR = KC + 4;                    // stride%64==4 -> conflict-free col reads
    __shared__ float Alds[16 * ASTR];
    __shared__ float Blds[KC * NCP];

    const int rowBase = blockIdx.x * 16;
    const int lane = threadIdx.x & 31;
    const int wave = threadIdx.x >> 5;
    const int lm   = lane & 15;
    const int kb   = (lane >> 4) << 1;              // lanes 16-31 hold K+2,K+3
    const int col  = wave * 16 + lm;

    v8f acc = {};
    for (int kc = 0; kc < KDIM; kc += KC) {
        for (int idx = threadIdx.x; idx < 16 * KC; idx += blockDim.x) {
            int r = idx / KC, c = idx - r * KC;
            Alds[r * ASTR + c] = A[(size_t)(rowBase + r) * KDIM + kc + c];
        }
        for (int idx = threadIdx.x; idx < KC * NCP; idx += blockDim.x) {
            int k = idx / NCP, c = idx - k * NCP;
            Blds[idx] = (c < NCOLS) ? W[(size_t)(kc + k) * NCOLS + c] : 0.0f;
        }
        __syncthreads();
#pragma unroll
        for (int k0 = 0; k0 < KC; k0 += 4) {
            v2f a, b;
            a.x = Alds[lm * ASTR + k0 + kb];
            a.y = Alds[lm * ASTR + k0 + kb + 1];
            b.x = Blds[(k0 + kb) * NCP + col];
            b.y = Blds[(k0 + kb + 1) * NCP + col];
            acc = __builtin_amdgcn_wmma_f32_16x16x4_f32(false, a, false, b,
                                                        (short)0, acc, false, false);
        }
        __syncthreads();
    }

    if (col < NCOLS) {
        float bv = bias[col];
#pragma unroll
        for (int r = 0; r < 8; ++r) {
            int grow = rowBase + r + ((lane < 16) ? 0 : 8);   // C/D VGPR layout
            float v = acc[r] + bv;
            if (EPI == 1)      v = fmaxf(v, 0.0f);
            else if (EPI == 2) v = (v > 0.0f) ? (1.0f + v) : __expf(v);
            out[(size_t)grow * NCOLS + col] = v;
        }
    }
}

// ---------------- build fused state S = [M(640) | Kt(64)] ----------------
__global__ void k_outer(const float* __restrict__ Kt0, const float* __restrict__ Vb,
                        float* __restrict__ S) {
    int idx = blockIdx.x * blockDim.x + threadIdx.x;
    if (idx >= NNODES * 640) return;
    int n = idx / 640, r = idx - n * 640;
    int h = r / 160, rem = r - h * 160;
    int i = rem / 10, j = rem - i * 10;
    S[(size_t)n * STATE + r] = Kt0[n * 64 + h * 16 + i] * Vb[n * 40 + h * 10 + j];
}
__global__ void k_copy_kt(const float* __restrict__ Kt0, float* __restrict__ S) {
    int idx = blockIdx.x * blockDim.x + threadIdx.x;
    if (idx >= NNODES * 64) return;
    int n = idx >> 6, t = idx & 63;
    S[(size_t)n * STATE + 640 + t] = Kt0[idx];
}
__global__ void k_hidden_init(const float* __restrict__ Vb,
                              const float* __restrict__ hopwise,
                              float* __restrict__ hid) {
    int idx = blockIdx.x * blockDim.x + threadIdx.x;
    if (idx < NNODES * 40) hid[idx] = Vb[idx] * hopwise[0];
}

// ---------------- one hop of propagation ----------------
// Self-loop folded into destination init: S_next = dis^2 * S_cur   (float4)
__global__ void k_selfinit(const float* __restrict__ dis, const float4* __restrict__ Scur,
                           float4* __restrict__ Snext) {
    int idx = blockIdx.x * blockDim.x + threadIdx.x;
    if (idx >= NNODES * (STATE / 4)) return;
    int n = idx / (STATE / 4);
    float d = dis[n]; d *= d;
    float4 s = Scur[idx];
    Snext[idx] = make_float4(d * s.x, d * s.y, d * s.z, d * s.w);
}
// One block per edge: float4 gather + hardware f32 atomics into L2-resident state
__global__ void k_scatter(const int* __restrict__ ei, const float* __restrict__ dis,
                          const float* __restrict__ Scur, float* __restrict__ Snext) {
    int e   = blockIdx.x;
    int row = ei[e];
    int col = ei[NEDGES + e];
    float nrm = dis[row] * dis[col];
    const float4* s = (const float4*)(Scur + (size_t)row * STATE);
    float*        d = Snext + (size_t)col * STATE;
    int f4 = threadIdx.x;                 // blockDim = 192, STATE/4 = 176
    if (f4 < STATE / 4) {
        float4 v = s[f4];
        atomicAddF(d + 4 * f4 + 0, nrm * v.x);
        atomicAddF(d + 4 * f4 + 1, nrm * v.y);
        atomicAddF(d + 4 * f4 + 2, nrm * v.z);
        atomicAddF(d + 4 * f4 + 3, nrm * v.w);
    }
}
// hidden[n,h,j] += gamma_h * (Q.M)/(Q.Kt + eps); head-axis softmax of temp inline
__global__ void k_update(const float* __restrict__ Q, const float* __restrict__ S,
                         const float* __restrict__ hopwise, const float* __restrict__ temp,
                         float* __restrict__ hid, int hop) {
    int idx = blockIdx.x * blockDim.x + threadIdx.x;
    if (idx >= NNODES * 40) return;
    int n = idx / 40, r = idx - n * 40;
    int h = r / 10,  j = r - h * 10;
    const float* q  = Q + (size_t)n * 64 + h * 16;
    const float* kt = S + (size_t)n * STATE + 640 + h * 16;
    const float* m  = S + (size_t)n * STATE + h * 160 + j;
    float cd = 1e-5f, hh = 0.0f;
#pragma unroll
    for (int i = 0; i < 16; ++i) {
        cd = fmaf(q[i], kt[i], cd);
        hh = fmaf(q[i], m[i * 10], hh);
    }
    int k = hop + 1;
    float den = __expf(temp[0 + k]) + __expf(temp[4 + k]) +
                __expf(temp[8 + k]) + __expf(temp[12 + k]);
    float gamma = hopwise[k] * __expf(temp[h * 4 + k]) / den;
    hid[idx] += gamma * hh / cd;
}

extern "C" void kernel_launch(void* const* d_in, const int* in_sizes, int n_in,
                              void* d_out, int out_size, void* d_ws, size_t ws_size,
                              hipStream_t stream) {
    const float* node_feat = (const float*)d_in[0];
    const int*   ei        = (const int*)  d_in[1];
    const float* W_in  = (const float*)d_in[2];
    const float* b_in  = (const float*)d_in[3];
    const float* Wq    = (const float*)d_in[4];
    const float* bq    = (const float*)d_in[5];
    const float* Wk    = (const float*)d_in[6];
    const float* bk    = (const float*)d_in[7];
    const float* Wv    = (const float*)d_in[8];
    const float* bv    = (const float*)d_in[9];
    const float* W_out = (const float*)d_in[10];
    const float* b_out = (const float*)d_in[11];
    const float* hopw  = (const float*)d_in[12];
    const float* temp  = (const float*)d_in[13];
    float* out = (float*)d_out;

    const int N = NNODES, E = NEDGES;
    float* ws  = (float*)d_ws;
    float* dis = ws;                 ws += N;
    float* x   = ws;                 ws += (size_t)N * 64;
    float* Qb  = ws;                 ws += (size_t)N * 64;
    float* Kt0 = ws;                 ws += (size_t)N * 64;
    float* Vb  = ws;                 ws += (size_t)N * 40;
    float* hid = ws;                 ws += (size_t)N * 40;
    float* S0  = ws;                 ws += (size_t)N * STATE;
    float* S1  = ws;                 /* total ~135 MB */

    // symmetric-normalized weights: dis = 1/sqrt(deg), deg counts col incl. self loop
    k_fill <<<(N + 255) / 256, 256, 0, stream>>>(dis, 1.0f, N);
    k_count<<<(E + 255) / 256, 256, 0, stream>>>(ei + E, dis, E);
    k_rsqrt<<<(N + 255) / 256, 256, 0, stream>>>(dis, N);

    // dense projections via f32 WMMA (compile-time shapes)
    k_gemm_wmma<256, 64, 1><<<N / 16, 128, 0, stream>>>(node_feat, W_in, b_in, x); // relu
    k_gemm_wmma< 64, 64, 2><<<N / 16, 128, 0, stream>>>(x, Wq, bq, Qb);            // 1+elu
    k_gemm_wmma< 64, 64, 2><<<N / 16, 128, 0, stream>>>(x, Wk, bk, Kt0);           // 1+elu
    k_gemm_wmma< 64, 40, 0><<<N / 16,  96, 0, stream>>>(x, Wv, bv, Vb);            // none

    k_outer      <<<(N * 640 + 255) / 256, 256, 0, stream>>>(Kt0, Vb, S0);
    k_copy_kt    <<<(N * 64  + 255) / 256, 256, 0, stream>>>(Kt0, S0);
    k_hidden_init<<<(N * 40  + 255) / 256, 256, 0, stream>>>(Vb, hopw, hid);

    float* Scur = S0; float* Snext = S1;
    for (int hop = 0; hop < 3; ++hop) {
        k_selfinit<<<(N * (STATE / 4) + 255) / 256, 256, 0, stream>>>(dis, (const float4*)Scur,
                                                                      (float4*)Snext);
        k_scatter <<<E, 192, 0, stream>>>(ei, dis, Scur, Snext);
        k_update  <<<(N * 40 + 255) / 256, 256, 0, stream>>>(Qb, Snext, hopw, temp, hid, hop);
        float* t = Scur; Scur = Snext; Snext = t;
    }

    // out = hidden[N x 40] @ W_out[40 x 10] + b_out   (WMMA, 1 wave/block)
    k_gemm_wmma<40, 10, 0><<<N / 16, 32, 0, stream>>>(hid, W_out, b_out, out);
}